// GlobalGraphNet_77360950936270
// MI455X (gfx1250) — compile-verified
//
#include <hip/hip_runtime.h>
#include <hip/hip_bf16.h>

#define Nn   38332
#define Ee   1200000
#define Cc   64
#define POI  38333
#define NLAY 5

typedef float v2f __attribute__((ext_vector_type(2)));
typedef float v8f __attribute__((ext_vector_type(8)));

static __device__ __forceinline__ float lrelu01(float v) { return v > 0.f ? v : 0.01f * v; }
static __device__ __forceinline__ float lrelu02(float v) { return v > 0.f ? v : 0.2f  * v; }

// order-preserving float<->uint mapping (for atomicMax on signed floats)
static __device__ __forceinline__ unsigned fOrd(float f) {
    unsigned u = __float_as_uint(f);
    return (u & 0x80000000u) ? ~u : (u | 0x80000000u);
}
static __device__ __forceinline__ float ordF(unsigned o) {
    return (o & 0x80000000u) ? __uint_as_float(o & 0x7fffffffu) : __uint_as_float(~o);
}
#define ORD_NEG_INF 0x007FFFFFu  /* fOrd(-inf) */

// ---------------------------------------------------------------- utilities
__global__ void k_fill_f(float* p, float v, int n) {
    int i = blockIdx.x * blockDim.x + threadIdx.x;
    if (i < n) p[i] = v;
}
__global__ void k_fill_u(unsigned* p, unsigned v, int n) {
    int i = blockIdx.x * blockDim.x + threadIdx.x;
    if (i < n) p[i] = v;
}

// ---------------------------------------------------------------- GCN norm prep
__global__ void k_scatter_deg(const int* __restrict__ dst, const float* __restrict__ w,
                              float* __restrict__ deg) {
    int e = blockIdx.x * blockDim.x + threadIdx.x;
    if (e < Ee) atomicAdd(&deg[dst[e]], w[e]);
}
__global__ void k_dinv(const float* __restrict__ deg, float* __restrict__ dinv) {
    int n = blockIdx.x * blockDim.x + threadIdx.x;
    if (n < Nn) {
        float d = deg[n] + 1.0f;                 // self-loop weight 1
        dinv[n] = (d > 0.f) ? __frsqrt_rn(fmaxf(d, 1e-12f)) : 0.f;
    }
}
__global__ void k_norm(const int* __restrict__ src, const int* __restrict__ dst,
                       const float* __restrict__ w, const float* __restrict__ dinv,
                       float* __restrict__ normw) {
    int e = blockIdx.x * blockDim.x + threadIdx.x;
    if (e < Ee) normw[e] = dinv[src[e]] * w[e] * dinv[dst[e]];
}

// ---------------------------------------------------------------- WMMA GEMMs (fp32)
// [N,64] @ [64,64] -> [N,64].  256 threads = 8 waves; each wave owns one 16x16 tile
// (2 row-tiles x 4 col-tiles per block).  W staged to LDS transposed (stride 66:
// conflict-free, even => ds_load_b64-aligned).  A fragments preloaded so the 16
// v_wmma_f32_16x16x4_f32 issue back-to-back.
#define WT 66
__global__ void __launch_bounds__(256) k_gemm64(const float* __restrict__ X,
                                                const float* __restrict__ W,
                                                float* __restrict__ H) {
    __shared__ float Ws[64 * WT];
    int tid = threadIdx.x;
#pragma unroll
    for (int i = 0; i < 16; ++i) {
        int idx = tid + i * 256;           // coalesced global read
        int kr = idx >> 6, nc = idx & 63;
        Ws[nc * WT + kr] = W[idx];         // transposed store, bank stride 2
    }
    __syncthreads();

    int wave = tid >> 5, lane = tid & 31;
    int half = lane >> 4, l = lane & 15;
    int mBase = (blockIdx.x * 2 + (wave >> 2)) * 16;
    int nCol  = (wave & 3) * 16 + l;
    int m  = mBase + l;
    int mc = min(m, Nn - 1);               // clamp: OOB rows masked at store

    const float2* Xp = (const float2*)(X + (size_t)mc * 64) + half;
    const float2* Bp = (const float2*)(Ws + nCol * WT);

    float2 a[16];
#pragma unroll
    for (int k = 0; k < 16; ++k) a[k] = Xp[k * 2];   // X[mc][4k + 2*half .. +1]

    v8f acc = {0.f, 0.f, 0.f, 0.f, 0.f, 0.f, 0.f, 0.f};
#pragma unroll
    for (int k = 0; k < 16; ++k) {
        float2 bv = Bp[k * 2 + half];                // W[4k+2*half .. +1][nCol]
        v2f A = {a[k].x, a[k].y};
        v2f B = {bv.x, bv.y};
        acc = __builtin_amdgcn_wmma_f32_16x16x4_f32(false, A, false, B, (short)0, acc,
                                                    false, false);
    }
#pragma unroll
    for (int v = 0; v < 8; ++v) {
        int row = mBase + v + half * 8;
        if (row < Nn) H[(size_t)row * 64 + nCol] = acc[v];
    }
}

// fused embedding-gather + [N,403]@[403,64] projection.
// K split at segment boundaries (300/400/403) => branch-free contiguous float2
// streams for the poi/cat segments; row index clamped instead of predicated.
__global__ void __launch_bounds__(32) k_gemm_in(const int* __restrict__ poi_idx,
                                                const int* __restrict__ cat_idx,
                                                const float* __restrict__ feat,
                                                const float* __restrict__ poi_emb,
                                                const float* __restrict__ cat_emb,
                                                const float* __restrict__ W,
                                                float* __restrict__ H) {
    int lane = threadIdx.x;
    int half = lane >> 4, l = lane & 15;
    int mBase = blockIdx.x * 16;
    int nCol  = blockIdx.y * 16 + l;
    int m  = mBase + l;
    int mc = min(m, Nn - 1);
    int pi = poi_idx[mc];
    int ci = cat_idx[mc];

    const float2* pe = (const float2*)(poi_emb + (size_t)pi * 300) + half;
    const float2* ce = (const float2*)(cat_emb + (size_t)ci * 100) + half;
    const float*  Wb = W + nCol;

    v8f acc = {0.f, 0.f, 0.f, 0.f, 0.f, 0.f, 0.f, 0.f};

    // K segment [0,300): poi embedding, 75 steps of K=4
    for (int k = 0; k < 75; ++k) {
        float2 av = pe[k * 2];
        int ka = k * 4 + half * 2;
        v2f A = {av.x, av.y};
        v2f B = {Wb[ka * 64], Wb[(ka + 1) * 64]};
        acc = __builtin_amdgcn_wmma_f32_16x16x4_f32(false, A, false, B, (short)0, acc,
                                                    false, false);
    }
    // K segment [300,400): cat embedding, 25 steps
    for (int k = 0; k < 25; ++k) {
        float2 av = ce[k * 2];
        int ka = 300 + k * 4 + half * 2;
        v2f A = {av.x, av.y};
        v2f B = {Wb[ka * 64], Wb[(ka + 1) * 64]};
        acc = __builtin_amdgcn_wmma_f32_16x16x4_f32(false, A, false, B, (short)0, acc,
                                                    false, false);
    }
    // K tail [400,404): feat (3 cols) + zero pad; one uniform-per-half branch
    {
        v2f A, B;
        if (half == 0) {
            A.x = feat[mc * 3 + 0]; A.y = feat[mc * 3 + 1];
            B.x = Wb[400 * 64];     B.y = Wb[401 * 64];
        } else {
            A.x = feat[mc * 3 + 2]; A.y = 0.f;
            B.x = Wb[402 * 64];     B.y = 0.f;
        }
        acc = __builtin_amdgcn_wmma_f32_16x16x4_f32(false, A, false, B, (short)0, acc,
                                                    false, false);
    }
#pragma unroll
    for (int v = 0; v < 8; ++v) {
        int row = mBase + v + half * 8;
        if (row < Nn) H[(size_t)row * 64 + nCol] = acc[v];
    }
}

// ---------------------------------------------------------------- GCN conv pieces
__global__ void k_conv_init(const float* __restrict__ h, const float* __restrict__ dinv,
                            const float* __restrict__ b, float* __restrict__ t) {
    int i = blockIdx.x * blockDim.x + threadIdx.x;
    if (i < Nn * Cc) {
        int n = i >> 6, c = i & 63;
        float dv = dinv[n];
        t[i] = h[i] * dv * dv + b[c];
    }
}
__global__ void k_conv_scatter(const int* __restrict__ src, const int* __restrict__ dst,
                               const float* __restrict__ normw, const float* __restrict__ h,
                               float* __restrict__ t) {
    long long i = (long long)blockIdx.x * blockDim.x + threadIdx.x;
    if (i < (long long)Ee * 16) {
        int e = (int)(i >> 4), q = (int)(i & 15);
        int s = src[e], d = dst[e];
        float nm = normw[e];
        float4 hv = reinterpret_cast<const float4*>(h)[s * 16 + q];
        float* tp = &t[d * 64 + q * 4];
        atomicAdd(tp + 0, hv.x * nm);
        atomicAdd(tp + 1, hv.y * nm);
        atomicAdd(tp + 2, hv.z * nm);
        atomicAdd(tp + 3, hv.w * nm);
    }
}
__global__ void k_lrelu_set(const float* __restrict__ t, float* __restrict__ x) {
    int i = blockIdx.x * blockDim.x + threadIdx.x;
    if (i < Nn * Cc) x[i] = lrelu01(t[i]);
}

// ---------------------------------------------------------------- GraphNorm
#define STAT_ROWS 256
__global__ void __launch_bounds__(64) k_stats(const float* __restrict__ t,
                                              float* __restrict__ stats) {
    int c  = threadIdx.x;
    int r0 = blockIdx.x * STAT_ROWS;
    int r1 = min(r0 + STAT_ROWS, Nn);
    float s = 0.f, s2 = 0.f;
    for (int r = r0; r < r1; ++r) {
        float v = t[r * 64 + c];
        s += v; s2 += v * v;
    }
    atomicAdd(&stats[c], s);
    atomicAdd(&stats[64 + c], s2);
}
__global__ void k_gn_apply(const float* __restrict__ t, const float* __restrict__ stats,
                           const float* __restrict__ gw, const float* __restrict__ gb,
                           const float* __restrict__ ga, float* __restrict__ x) {
    int i = blockIdx.x * blockDim.x + threadIdx.x;
    if (i < Nn * Cc) {
        int c = i & 63;
        float invN = 1.0f / (float)Nn;
        float m  = stats[c] * invN;
        float m2 = stats[64 + c] * invN;
        float a  = ga[c];
        float var = m2 - 2.f * a * m * m + a * a * m * m;
        float g = (t[i] - a * m) * __frsqrt_rn(var + 1e-5f) * gw[c] + gb[c];
        x[i] += lrelu01(g);
    }
}

// ---------------------------------------------------------------- GAT pieces
__global__ void k_attn(const float* __restrict__ h, const float* __restrict__ as_v,
                       const float* __restrict__ ad_v, float* __restrict__ a_s,
                       float* __restrict__ a_d) {
    int n = blockIdx.x * blockDim.x + threadIdx.x;
    if (n < Nn) {
        float ss = 0.f, sd = 0.f;
        for (int c = 0; c < 64; ++c) {
            float hv = h[n * 64 + c];
            ss += hv * as_v[c];
            sd += hv * ad_v[c];
        }
        a_s[n] = ss; a_d[n] = sd;
    }
}
static __device__ __forceinline__ void gat_edge(int i, const int* src, const int* dst,
                                                int& s, int& d) {
    if (i < Ee) { s = src[i]; d = dst[i]; }
    else        { s = i - Ee; d = s; }       // self-loops appended
}
__global__ void k_gat_max(const int* __restrict__ src, const int* __restrict__ dst,
                          const float* __restrict__ a_s, const float* __restrict__ a_d,
                          unsigned* __restrict__ mmax) {
    int i = blockIdx.x * blockDim.x + threadIdx.x;
    if (i < Ee + Nn) {
        int s, d; gat_edge(i, src, dst, s, d);
        float e = lrelu02(a_s[s] + a_d[d]);
        atomicMax(&mmax[d], fOrd(e));
    }
}
__global__ void k_gat_exp(const int* __restrict__ src, const int* __restrict__ dst,
                          const float* __restrict__ a_s, const float* __restrict__ a_d,
                          const unsigned* __restrict__ mmax, float* __restrict__ exb,
                          float* __restrict__ z) {
    int i = blockIdx.x * blockDim.x + threadIdx.x;
    if (i < Ee + Nn) {
        int s, d; gat_edge(i, src, dst, s, d);
        float e = lrelu02(a_s[s] + a_d[d]);
        float ex = __expf(e - ordF(mmax[d]));
        exb[i] = ex;
        atomicAdd(&z[d], ex);
    }
}
__global__ void k_gat_scatter(const int* __restrict__ src, const int* __restrict__ dst,
                              const float* __restrict__ exb, const float* __restrict__ z,
                              const float* __restrict__ h, float* __restrict__ t) {
    long long i = (long long)blockIdx.x * blockDim.x + threadIdx.x;
    if (i < (long long)(Ee + Nn) * 16) {
        int e = (int)(i >> 4), q = (int)(i & 15);
        int s, d; gat_edge(e, src, dst, s, d);
        float al = exb[e] / (z[d] + 1e-16f);
        float4 hv = reinterpret_cast<const float4*>(h)[s * 16 + q];
        float* tp = &t[d * 64 + q * 4];
        atomicAdd(tp + 0, hv.x * al);
        atomicAdd(tp + 1, hv.y * al);
        atomicAdd(tp + 2, hv.z * al);
        atomicAdd(tp + 3, hv.w * al);
    }
}
__global__ void k_add_bias(float* __restrict__ t, const float* __restrict__ b) {
    int i = blockIdx.x * blockDim.x + threadIdx.x;
    if (i < Nn * Cc) t[i] += b[i & 63];
}

// ---------------------------------------------------------------- output GCN (C=1)
__global__ void k_outdot(const float* __restrict__ x, const float* __restrict__ Wout,
                         float* __restrict__ h1) {
    int n = blockIdx.x * blockDim.x + threadIdx.x;
    if (n < Nn) {
        float s = 0.f;
        for (int c = 0; c < 64; ++c) s += x[n * 64 + c] * Wout[c];
        h1[n] = s;
    }
}
__global__ void k_out_init(const float* __restrict__ h1, const float* __restrict__ dinv,
                           const float* __restrict__ bout, float* __restrict__ t1) {
    int n = blockIdx.x * blockDim.x + threadIdx.x;
    if (n < Nn) { float dv = dinv[n]; t1[n] = h1[n] * dv * dv + bout[0]; }
}
__global__ void k_out_scatter(const int* __restrict__ src, const int* __restrict__ dst,
                              const float* __restrict__ normw, const float* __restrict__ h1,
                              float* __restrict__ t1) {
    int e = blockIdx.x * blockDim.x + threadIdx.x;
    if (e < Ee) atomicAdd(&t1[dst[e]], h1[src[e]] * normw[e]);
}
__global__ void k_out_final(const float* __restrict__ t1, float* __restrict__ xf) {
    int n = blockIdx.x * blockDim.x + threadIdx.x;
    if (n < Nn) xf[n] = lrelu01(t1[n]);
}

// ---------------------------------------------------------------- final FCs (HBM-bound MV)
#define FC1_ROWS 128
__global__ void __launch_bounds__(128) k_fc1(const float* __restrict__ xf,
                                             const float* __restrict__ W1,
                                             float* __restrict__ h128) {
    int c  = threadIdx.x;
    int r0 = blockIdx.x * FC1_ROWS;
    int r1 = min(r0 + FC1_ROWS, Nn);
    float acc = 0.f;
    for (int r = r0; r < r1; ++r) {
        if (r + 16 < r1) __builtin_prefetch(&W1[(size_t)(r + 16) * 128 + c], 0, 1);
        acc += xf[r] * W1[(size_t)r * 128 + c];
    }
    atomicAdd(&h128[c], acc);
}
__global__ void k_fc2(const float* __restrict__ h128, const float* __restrict__ b1,
                      const float* __restrict__ W2, const float* __restrict__ b2,
                      float* __restrict__ out) {
    int j = blockIdx.x * blockDim.x + threadIdx.x;
    if (j < POI) {
        float acc = b2[j];
        for (int k = 0; k < 128; ++k) {
            float hk = fmaxf(h128[k] + b1[k], 0.f);
            acc += hk * W2[(size_t)k * POI + j];
        }
        out[j] = fmaxf(acc, 0.f);
    }
}

// ---------------------------------------------------------------- host driver
static inline int cdiv(long long a, long long b) { return (int)((a + b - 1) / b); }

extern "C" void kernel_launch(void* const* d_in, const int* in_sizes, int n_in,
                              void* d_out, int out_size, void* d_ws, size_t ws_size,
                              hipStream_t stream) {
    const int*   poi_idx = (const int*)d_in[0];
    const int*   cat_idx = (const int*)d_in[1];
    const float* feat    = (const float*)d_in[2];
    const int*   eidx    = (const int*)d_in[3];
    const float* weight  = (const float*)d_in[4];
    const float* poi_emb = (const float*)d_in[5];
    const float* cat_emb = (const float*)d_in[6];
    const float* W_in    = (const float*)d_in[7];
    const float* b_in    = (const float*)d_in[8];
    const float* gcn_W   = (const float*)d_in[9];
    const float* gcn_b   = (const float*)d_in[10];
    const float* gn_w    = (const float*)d_in[11];
    const float* gn_b    = (const float*)d_in[12];
    const float* gn_a    = (const float*)d_in[13];
    const float* gat_W   = (const float*)d_in[14];
    const float* gat_as  = (const float*)d_in[15];
    const float* gat_ad  = (const float*)d_in[16];
    const float* gat_b   = (const float*)d_in[17];
    const float* W_out   = (const float*)d_in[18];
    const float* b_out   = (const float*)d_in[19];
    const float* fc1_W   = (const float*)d_in[20];
    const float* fc1_b   = (const float*)d_in[21];
    const float* fc2_W   = (const float*)d_in[22];
    const float* fc2_b   = (const float*)d_in[23];
    float* out = (float*)d_out;

    const int* src = eidx;
    const int* dst = eidx + Ee;

    float* ws = (float*)d_ws;
    const size_t N64 = (size_t)Nn * 64;
    const size_t NP  = 38400;
    size_t off = 0;
    float*    x     = ws + off; off += N64;
    float*    h     = ws + off; off += N64;
    float*    t     = ws + off; off += N64;
    float*    deg   = ws + off; off += NP;
    float*    dinv  = ws + off; off += NP;
    float*    normw = ws + off; off += 1200128;
    float*    a_s   = ws + off; off += NP;
    float*    a_d   = ws + off; off += NP;
    float*    zb    = ws + off; off += NP;
    unsigned* mmax  = (unsigned*)(ws + off); off += NP;
    float*    exb   = ws + off; off += 1238400;
    float*    stats = ws + off; off += 128;
    float*    h1    = ws + off; off += NP;
    float*    t1    = ws + off; off += NP;
    float*    xf    = ws + off; off += NP;
    float*    h128  = ws + off; off += 128;

    const int B = 256;
    dim3 gemmGrid(cdiv(Nn, 32));          // k_gemm64: 2 row-tiles x 4 col-tiles / block
    dim3 gemmInGrid(cdiv(Nn, 16), 4);     // k_gemm_in: 1 wave per tile

    // ---- GCN normalization coefficients (shared by all 7 GCN convs)
    k_fill_f<<<cdiv(Nn, B), B, 0, stream>>>(deg, 0.f, Nn);
    k_scatter_deg<<<cdiv(Ee, B), B, 0, stream>>>(dst, weight, deg);
    k_dinv<<<cdiv(Nn, B), B, 0, stream>>>(deg, dinv);
    k_norm<<<cdiv(Ee, B), B, 0, stream>>>(src, dst, weight, dinv, normw);

    // ---- input projection (gather + WMMA) then GCN conv + lrelu -> x
    k_gemm_in<<<gemmInGrid, 32, 0, stream>>>(poi_idx, cat_idx, feat, poi_emb, cat_emb,
                                             W_in, h);
    k_conv_init<<<cdiv(Nn * Cc, B), B, 0, stream>>>(h, dinv, b_in, t);
    k_conv_scatter<<<cdiv((long long)Ee * 16, B), B, 0, stream>>>(src, dst, normw, h, t);
    k_lrelu_set<<<cdiv(Nn * Cc, B), B, 0, stream>>>(t, x);

    for (int i = 0; i < NLAY; ++i) {
        // ---- GCN block
        k_gemm64<<<gemmGrid, 256, 0, stream>>>(x, gcn_W + (size_t)i * 64 * 64, h);
        k_conv_init<<<cdiv(Nn * Cc, B), B, 0, stream>>>(h, dinv, gcn_b + i * 64, t);
        k_conv_scatter<<<cdiv((long long)Ee * 16, B), B, 0, stream>>>(src, dst, normw, h, t);
        k_fill_f<<<1, 128, 0, stream>>>(stats, 0.f, 128);
        k_stats<<<cdiv(Nn, STAT_ROWS), 64, 0, stream>>>(t, stats);
        k_gn_apply<<<cdiv(Nn * Cc, B), B, 0, stream>>>(t, stats, gn_w + i * 64,
                                                       gn_b + i * 64, gn_a + i * 64, x);
        // ---- GAT block
        k_gemm64<<<gemmGrid, 256, 0, stream>>>(x, gat_W + (size_t)i * 64 * 64, h);
        k_attn<<<cdiv(Nn, B), B, 0, stream>>>(h, gat_as + i * 64, gat_ad + i * 64, a_s, a_d);
        k_fill_f<<<cdiv(Nn, B), B, 0, stream>>>(zb, 0.f, Nn);
        k_fill_u<<<cdiv(Nn, B), B, 0, stream>>>(mmax, ORD_NEG_INF, Nn);
        k_fill_f<<<cdiv(Nn * Cc, B), B, 0, stream>>>(t, 0.f, Nn * Cc);
        k_gat_max<<<cdiv(Ee + Nn, B), B, 0, stream>>>(src, dst, a_s, a_d, mmax);
        k_gat_exp<<<cdiv(Ee + Nn, B), B, 0, stream>>>(src, dst, a_s, a_d, mmax, exb, zb);
        k_gat_scatter<<<cdiv((long long)(Ee + Nn) * 16, B), B, 0, stream>>>(src, dst, exb,
                                                                            zb, h, t);
        k_add_bias<<<cdiv(Nn * Cc, B), B, 0, stream>>>(t, gat_b + i * 64);
        k_fill_f<<<1, 128, 0, stream>>>(stats, 0.f, 128);
        k_stats<<<cdiv(Nn, STAT_ROWS), 64, 0, stream>>>(t, stats);
        k_gn_apply<<<cdiv(Nn * Cc, B), B, 0, stream>>>(t, stats, gn_w + i * 64,
                                                       gn_b + i * 64, gn_a + i * 64, x);
    }

    // ---- output GCN conv (C=1) + lrelu
    k_outdot<<<cdiv(Nn, B), B, 0, stream>>>(x, W_out, h1);
    k_out_init<<<cdiv(Nn, B), B, 0, stream>>>(h1, dinv, b_out, t1);
    k_out_scatter<<<cdiv(Ee, B), B, 0, stream>>>(src, dst, normw, h1, t1);
    k_out_final<<<cdiv(Nn, B), B, 0, stream>>>(t1, xf);

    // ---- final FCs
    k_fill_f<<<1, 128, 0, stream>>>(h128, 0.f, 128);
    k_fc1<<<cdiv(Nn, FC1_ROWS), 128, 0, stream>>>(xf, fc1_W, h128);
    k_fc2<<<cdiv(POI, B), B, 0, stream>>>(h128, fc1_b, fc2_W, fc2_b, out);
}